// HaarDWTEdgePrior_23579370455122
// MI455X (gfx1250) — compile-verified
//
#include <hip/hip_runtime.h>
#include <cstdint>

// Haar HL/LH/HH 2x2 stencil edge magnitude, reflect pad right/bottom.
// x: (8, 64, 256, 256) f32 -> out same shape.
//
// Memory-bound (256 MB traffic, ~11 us floor @ 23.3 TB/s). Strategy:
//  - async global->LDS DMA (CDNA5 GLOBAL_LOAD_ASYNC_TO_LDS_B128, ASYNCcnt)
//    stages a (32+1)x256 row tile per block; each interior element is read
//    from HBM exactly once.
//  - compute reads neighbors from LDS (ds_load_b128), stores NT to global.

typedef __attribute__((ext_vector_type(4))) float v4f;

#define IMG_W 256
#define IMG_H 256
#define TILE_H 32
#define TILE_ROWS (TILE_H + 1)          // +1 halo row
#define LDS_STRIDE 260                  // floats: 1040B pitch (16B aligned), bank-skewed
#define NTHREADS 256

__global__ __launch_bounds__(NTHREADS)
void haar_edge_kernel(const float* __restrict__ x, float* __restrict__ out) {
  __shared__ float tile[TILE_ROWS * LDS_STRIDE];

  const int tid   = threadIdx.x;
  const int plane = blockIdx.x >> 3;       // b*C + c  (8 tiles per 256-row plane)
  const int r0    = (blockIdx.x & 7) * TILE_H;
  const size_t plane_base = (size_t)plane * (IMG_H * IMG_W);

  // ---- Stage (TILE_H+1) x 256 floats into LDS via CDNA5 async copies.
  // 33 rows * 64 float4-chunks = 2112 chunks; 256 threads -> <=9 per thread.
  for (int i = tid; i < TILE_ROWS * (IMG_W / 4); i += NTHREADS) {
    const int row  = i >> 6;
    const int col  = (i & 63) << 2;        // float index within row, 16B aligned
    int gr = r0 + row;
    if (gr >= IMG_H) gr = IMG_H - 2;       // reflect bottom pad: xp[H] = x[H-2]
    const float* gptr = x + plane_base + (size_t)gr * IMG_W + col;
    // Low 32 bits of the generic (flat) LDS address are the LDS byte offset
    // (ISA 10.2, LDS aperture).
    unsigned lds_addr = (unsigned)(uintptr_t)(const void*)&tile[row * LDS_STRIDE + col];
    asm volatile("global_load_async_to_lds_b128 %0, %1, off"
                 :: "v"(lds_addr), "v"(gptr)
                 : "memory");
  }
  // Wait for this wave's async transfers, then make them visible workgroup-wide.
  asm volatile("s_wait_asynccnt 0" ::: "memory");
  __syncthreads();

  // ---- Compute: thread handles column chunk (tid&63)*4, rows hbase+4k.
  const int col   = (tid & 63) << 2;       // 0..252 step 4
  const int hbase = tid >> 6;              // 0..3

#pragma unroll
  for (int k = 0; k < TILE_H / 4; ++k) {
    const int h = hbase + 4 * k;           // 0..31, unique over the block
    const float* rc = &tile[h * LDS_STRIDE + col];
    const float* rn = &tile[(h + 1) * LDS_STRIDE + col];
    const v4f cur = *(const v4f*)rc;       // x[h][col..col+3]
    const v4f nxt = *(const v4f*)rn;       // x[h+1][col..col+3]
    float cur4, nxt4;                      // x[.][col+4] with right reflect
    if (col == IMG_W - 4) {                // xp[..,256] = x[..,254] = rc[2]
      cur4 = rc[2];
      nxt4 = rn[2];
    } else {
      cur4 = rc[4];
      nxt4 = rn[4];
    }

    const float a[4] = {cur.x, cur.y, cur.z, cur.w};
    const float b[4] = {cur.y, cur.z, cur.w, cur4};
    const float c[4] = {nxt.x, nxt.y, nxt.z, nxt.w};
    const float d[4] = {nxt.y, nxt.z, nxt.w, nxt4};

    v4f o;
#pragma unroll
    for (int j = 0; j < 4; ++j) {
      // hl=.5(a+b-c-d), lh=.5(a-b+c-d), hh=.5(a-b-c+d)
      // With p=a-d, q=b-c, r=.5((a+d)-(b+c)):  hl^2+lh^2 = .5(p^2+q^2), hh=r.
      const float p = a[j] - d[j];
      const float q = b[j] - c[j];
      const float r = 0.5f * ((a[j] + d[j]) - (b[j] + c[j]));
      const float s = fmaf(p, p, q * q);
      const float t = fmaf(r, r, 0.5f * s);
      const float v = __builtin_amdgcn_sqrtf(t);
      if (j == 0) o.x = v; else if (j == 1) o.y = v; else if (j == 2) o.z = v; else o.w = v;
    }

    float* optr = out + plane_base + (size_t)(r0 + h) * IMG_W + col;
    __builtin_nontemporal_store(o, (v4f*)optr);   // th:TH_STORE_NT, keep L2 for halos
  }
}

extern "C" void kernel_launch(void* const* d_in, const int* in_sizes, int n_in,
                              void* d_out, int out_size, void* d_ws, size_t ws_size,
                              hipStream_t stream) {
  const float* x = (const float*)d_in[0];
  float* out = (float*)d_out;
  // 8 batches * 64 channels * (256/32) row-tiles = 4096 blocks
  const int nblocks = 8 * 64 * (IMG_H / TILE_H);
  hipLaunchKernelGGL(haar_edge_kernel, dim3(nblocks), dim3(NTHREADS), 0, stream, x, out);
}